// MetaFeatureExtractor_14628658610371
// MI455X (gfx1250) — compile-verified
//
#include <hip/hip_runtime.h>
#include <hip/hip_bf16.h>
#include <math.h>

typedef __attribute__((ext_vector_type(2))) float v2f;
typedef __attribute__((ext_vector_type(8))) float v8f;

#define T_DIM   2048
#define C_DIM   64
#define TSLICES 4          // T-parallel waves per channel group
#define CGROUPS 4          // 64 channels / 16-wide WMMA N dimension
#define WAVES   (TSLICES * CGROUPS)
#define BLOCK   (WAVES * 32)

// One workgroup per batch. Each wave owns (channel-group, T-slice) and
// reduces its 512-timestep slab with V_WMMA_F32_16X16X4_F32:
//   sum   = ones(16x4) @ x_tile(4x16)   accumulated in C/D
//   sumsq = ones(16x4) @ (x*x)_tile     accumulated in C/D
// Max/min tracked in VALU (free: kernel is HBM-bound).
__global__ __launch_bounds__(BLOCK)
void meta_feature_kernel(const float* __restrict__ x, float* __restrict__ out) {
    const int b    = blockIdx.x;
    const int tid  = threadIdx.x;
    const int wave = tid >> 5;
    const int lane = tid & 31;
    const int half = lane >> 4;   // which pair of timesteps this lane feeds
    const int ln   = lane & 15;   // channel within group == WMMA N index
    const int cg   = wave & (CGROUPS - 1);
    const int ts   = wave >> 2;

    const int channel = cg * 16 + ln;
    const float* __restrict__ xb = x + (size_t)b * T_DIM * C_DIM;

    const int slice   = T_DIM / TSLICES;        // 512
    const int t_begin = ts * slice;
    const int t_end   = t_begin + slice;

    v8f acc_s = {};   // running column sums      (every row of D identical)
    v8f acc_q = {};   // running column sum-of-sq
    v2f ones;  ones.x = 1.0f; ones.y = 1.0f;    // A = 16x4 all-ones

    float vmax = -INFINITY;
    float vmin =  INFINITY;

    for (int t0 = t_begin; t0 < t_end; t0 += 4) {
        // B layout for 4x16 f32: VGPR0 = K {0 | 2}, VGPR1 = K {1 | 3}
        const int r0 = t0 + half * 2;
        const float* p = xb + (size_t)r0 * C_DIM + channel;
        const float b0 = __builtin_nontemporal_load(p);          // row r0
        const float b1 = __builtin_nontemporal_load(p + C_DIM);  // row r0+1

        v2f bv; bv.x = b0;      bv.y = b1;
        v2f qv; qv.x = b0 * b0; qv.y = b1 * b1;

        acc_s = __builtin_amdgcn_wmma_f32_16x16x4_f32(
            false, ones, false, bv, (short)0, acc_s, false, false);
        acc_q = __builtin_amdgcn_wmma_f32_16x16x4_f32(
            false, ones, false, qv, (short)0, acc_q, false, false);

        vmax = fmaxf(vmax, fmaxf(b0, b1));
        vmin = fminf(vmin, fminf(b0, b1));
    }

    // The two lane-halves saw disjoint timestep phases: merge with xor-16.
    vmax = fmaxf(vmax, __shfl_xor(vmax, 16, 32));
    vmin = fminf(vmin, __shfl_xor(vmin, 16, 32));

    __shared__ float s_sum[WAVES * 16];
    __shared__ float s_sq [WAVES * 16];
    __shared__ float s_max[WAVES * 16];
    __shared__ float s_min[WAVES * 16];

    // acc[0] = D row 0 = full column sum for channel (lane&15); halves duplicate.
    if (half == 0) {
        s_sum[wave * 16 + ln] = acc_s[0];
        s_sq [wave * 16 + ln] = acc_q[0];
        s_max[wave * 16 + ln] = vmax;
        s_min[wave * 16 + ln] = vmin;
    }
    __syncthreads();

    if (tid < C_DIM) {
        const int c   = tid;
        const int cg2 = c >> 4;
        const int ln2 = c & 15;
        float sum = 0.0f, sq = 0.0f, mx = -INFINITY, mn = INFINITY;
        #pragma unroll
        for (int s = 0; s < TSLICES; ++s) {
            const int idx = (s * CGROUPS + cg2) * 16 + ln2;
            sum += s_sum[idx];
            sq  += s_sq [idx];
            mx   = fmaxf(mx, s_max[idx]);
            mn   = fminf(mn, s_min[idx]);
        }
        const float mean = sum / (float)T_DIM;
        float var = (sq - sum * mean) / (float)(T_DIM - 1);   // Bessel ddof=1
        var = fmaxf(var, 0.0f);
        const float stddev = sqrtf(var);

        const float first = xb[c];
        const float last  = xb[(size_t)(T_DIM - 1) * C_DIM + c];
        const float slope = (last - first) / (float)(T_DIM - 1);

        float* __restrict__ ob = out + (size_t)b * 5 * C_DIM;
        ob[0 * C_DIM + c] = mean;
        ob[1 * C_DIM + c] = stddev;
        ob[2 * C_DIM + c] = mx;
        ob[3 * C_DIM + c] = mn;
        ob[4 * C_DIM + c] = slope;
    }
}

extern "C" void kernel_launch(void* const* d_in, const int* in_sizes, int n_in,
                              void* d_out, int out_size, void* d_ws, size_t ws_size,
                              hipStream_t stream) {
    const float* x  = (const float*)d_in[0];
    float* out      = (float*)d_out;
    const int batch = in_sizes[0] / (T_DIM * C_DIM);   // 256
    meta_feature_kernel<<<batch, BLOCK, 0, stream>>>(x, out);
}